// GraphSAGELayer_18915035971927
// MI455X (gfx1250) — compile-verified
//
#include <hip/hip_runtime.h>
#include <math.h>

// GraphSAGE layer for MI455X (gfx1250, wave32).
// Pipeline:
//   1) agg_max_kernel  : masked max-aggregation (memory-bound; adjacency read
//                        once, staged through LDS, uniform-branch skip of the
//                        95% zero entries; features stay L2-resident).
//   2) sage_gemm_kernel: fused  relu(X*Wself + G*Wneigh + b)  via
//                        V_WMMA_F32_16X16X4_F32 (fp32 tensor path), then
//                        row-wise L2 normalization with an LDS ds_add_f32
//                        cross-wave reduction.

#define EPSV 1e-7f
#define NEGV -1000000000.0f

constexpr int Bk = 4;
constexpr int Uk = 1024;
constexpr int Nk = 512;
constexpr int Dk = 64;
constexpr int VT = Uk + Nk;   // 1536, adjacency side length

typedef float v2f __attribute__((ext_vector_type(2)));
typedef float v8f __attribute__((ext_vector_type(8)));

// ---------------------------------------------------------------------------
// Kernel 1: masked max aggregation.
//   agg[b, row, d] = max over c in [0,COLS) with adj[b, row_off+row, col_off+c] > 0
//                    of feat[b, c, d];   NEGV if no neighbor.
// Block = 64 threads (2 waves), one block per (b, row). Thread d owns dim d.
// Adjacency row chunk staged in LDS so all 64 lanes share one coalesced read.
// ---------------------------------------------------------------------------
template <int COLS, int ROWS>
__global__ __launch_bounds__(64)
void agg_max_kernel(const float* __restrict__ adj,   // (B, VT, VT)
                    const float* __restrict__ feat,  // (B, COLS, 64)
                    float* __restrict__ agg,         // (B, ROWS, 64)
                    int row_off, int col_off)
{
    __shared__ float s_adj[64];

    const int gr  = blockIdx.x;            // 0 .. B*ROWS-1
    const int b   = gr / ROWS;
    const int row = gr % ROWS;
    const int tid = threadIdx.x;           // 0..63 : feature dim

    const float* adj_row = adj + ((size_t)b * VT + row_off + row) * VT + col_off;
    const float* featb   = feat + (size_t)b * COLS * Dk;

    float acc = NEGV;

    for (int c0 = 0; c0 < COLS; c0 += 64) {
        __syncthreads();                   // protect s_adj reuse (WAR)
        s_adj[tid] = adj_row[c0 + tid];
        __syncthreads();
#pragma unroll 8
        for (int j = 0; j < 64; ++j) {
            // Uniform across the block -> scalar branch, skips 95% of cols.
            if (s_adj[j] > 0.0f) {
                acc = fmaxf(acc, featb[(size_t)(c0 + j) * Dk + tid]);
            }
        }
    }

    agg[(size_t)gr * Dk + tid] = acc;
}

// ---------------------------------------------------------------------------
// Kernel 2: fused dual-GEMM + bias + ReLU + row L2-normalize.
//   out(R,64) = l2norm_rows( relu( X(R,64)*Wself(64,64) + G(R,64)*Wneigh(64,64) + b ) )
// 256 threads = 8 waves per block; wave w -> 16x16 tile (row_tile = w/4,
// col_tile = w%4); block covers 32 rows x 64 cols. K=64 -> 16 WMMA steps per
// matrix, both accumulate into the same v8f C fragment.
// ---------------------------------------------------------------------------
__global__ __launch_bounds__(256)
void sage_gemm_kernel(const float* __restrict__ X,      // (R, 64)
                      const float* __restrict__ G,      // (R, 64)
                      const float* __restrict__ Wself,  // (64, 64) row-major
                      const float* __restrict__ Wneigh, // (64, 64) row-major
                      const float* __restrict__ bias,   // (64)
                      float* __restrict__ out)          // (R, 64)
{
    __shared__ float rowsq[32];

    const int tid      = threadIdx.x;
    const int wave     = tid >> 5;         // 0..7
    const int lane     = tid & 31;
    const int lhi      = lane >> 4;        // 0: K pair {0,1}; 1: K pair {2,3}
    const int llo      = lane & 15;
    const int row_tile = wave >> 2;        // 0..1
    const int col_tile = wave & 3;         // 0..3

    const int row_base = blockIdx.x * 32 + row_tile * 16;
    const int col      = col_tile * 16 + llo;

    if (tid < 32) rowsq[tid] = 0.0f;
    __syncthreads();

    // A fragment rows: M = llo for both lane halves (halves differ in K).
    const float* xrow = X + (size_t)(row_base + llo) * Dk;
    const float* grow = G + (size_t)(row_base + llo) * Dk;

    v8f c = {0.f, 0.f, 0.f, 0.f, 0.f, 0.f, 0.f, 0.f};

#pragma unroll
    for (int kk = 0; kk < Dk; kk += 4) {
        const int ka = kk + 2 * lhi;
        v2f a;  a.x  = xrow[ka];
                a.y  = xrow[ka + 1];
        v2f bf; bf.x = Wself[(size_t)ka * Dk + col];
                bf.y = Wself[(size_t)(ka + 1) * Dk + col];
        c = __builtin_amdgcn_wmma_f32_16x16x4_f32(false, a, false, bf,
                                                  (short)0, c, false, false);
    }
#pragma unroll
    for (int kk = 0; kk < Dk; kk += 4) {
        const int ka = kk + 2 * lhi;
        v2f a;  a.x  = grow[ka];
                a.y  = grow[ka + 1];
        v2f bf; bf.x = Wneigh[(size_t)ka * Dk + col];
                bf.y = Wneigh[(size_t)(ka + 1) * Dk + col];
        c = __builtin_amdgcn_wmma_f32_16x16x4_f32(false, a, false, bf,
                                                  (short)0, c, false, false);
    }

    // Bias (per-column) + ReLU, then per-row sum of squares via LDS atomics.
    const float bv = bias[col];
    float d[8];
#pragma unroll
    for (int r = 0; r < 8; ++r) {
        float v = c[r] + bv;
        d[r] = v > 0.0f ? v : 0.0f;
    }
#pragma unroll
    for (int r = 0; r < 8; ++r) {
        const int rowloc = row_tile * 16 + r + 8 * lhi;  // 0..31 in block
        atomicAdd(&rowsq[rowloc], d[r] * d[r]);          // ds_add_f32
    }
    __syncthreads();

#pragma unroll
    for (int r = 0; r < 8; ++r) {
        const int rowloc = row_tile * 16 + r + 8 * lhi;
        const float inv  = 1.0f / (sqrtf(rowsq[rowloc]) + EPSV);
        out[(size_t)(blockIdx.x * 32 + rowloc) * Dk + col] = d[r] * inv;
    }
}

// ---------------------------------------------------------------------------
extern "C" void kernel_launch(void* const* d_in, const int* in_sizes, int n_in,
                              void* d_out, int out_size, void* d_ws, size_t ws_size,
                              hipStream_t stream)
{
    (void)in_sizes; (void)n_in; (void)out_size; (void)ws_size;

    const float* user_features = (const float*)d_in[0]; // (B, U, 64)
    const float* news_features = (const float*)d_in[1]; // (B, N, 64)
    const float* adj           = (const float*)d_in[2]; // (B, 1536, 1536)
    const float* W_user_self   = (const float*)d_in[3]; // (64, 64)
    const float* W_user_neigh  = (const float*)d_in[4]; // (64, 64)
    const float* b_user        = (const float*)d_in[5]; // (64)
    const float* W_news_self   = (const float*)d_in[6]; // (64, 64)
    const float* W_news_neigh  = (const float*)d_in[7]; // (64, 64)
    const float* b_news        = (const float*)d_in[8]; // (64)

    float* out_users = (float*)d_out;                        // (B*U, 64)
    float* out_news  = out_users + (size_t)Bk * Uk * Dk;     // (B*N, 64)

    float* agg_users = (float*)d_ws;                         // news_for_users (B*U, 64)
    float* agg_news  = agg_users + (size_t)Bk * Uk * Dk;     // users_for_news (B*N, 64)

    // 1) Max-aggregation (memory-bound: ~16 MB adjacency traffic total).
    //    users aggregate news:  adj[:, :U, U:]  x  news_features
    agg_max_kernel<Nk, Uk><<<Bk * Uk, 64, 0, stream>>>(
        adj, news_features, agg_users, /*row_off=*/0, /*col_off=*/Uk);
    //    news aggregate users:  adj[:, U:, :U]  x  user_features
    agg_max_kernel<Uk, Nk><<<Bk * Nk, 64, 0, stream>>>(
        adj, user_features, agg_news, /*row_off=*/Uk, /*col_off=*/0);

    // 2) Fused WMMA GEMM + bias + ReLU + L2-normalize.
    sage_gemm_kernel<<<(Bk * Uk) / 32, 256, 0, stream>>>(
        user_features, agg_users, W_user_self, W_user_neigh, b_user, out_users);
    sage_gemm_kernel<<<(Bk * Nk) / 32, 256, 0, stream>>>(
        news_features, agg_news, W_news_self, W_news_neigh, b_news, out_news);
}